// Joiner_12249246729035
// MI455X (gfx1250) — compile-verified
//
#include <hip/hip_runtime.h>

typedef __attribute__((ext_vector_type(2))) float v2f;
typedef __attribute__((ext_vector_type(4))) float v4f;
typedef __attribute__((ext_vector_type(8))) float v8f;

// C[m,n] = sum_k A[m,k] * W[n, woff + k]  (+ bias[n])
// A: row-major M x lda. W: row-major, row length ldw. C: row-major M x N.
// One wave per 16x16 output tile, V_WMMA_F32_16X16X4_F32 over K.
// A-frag: lane(0-15)=row m0+r holds K={k,k+1}; lanes 16-31 hold K={k+2,k+3}.
// B-frag (B=W^T): lane(0-15)=col n0+r holds W[n, k..k+1]; upper half k+2..k+3.
// D layout: lane holds column n0+(l&15); VGPR i = row m0 + i + 8*(l>>4).
__global__ __launch_bounds__(256) void gemm_wmma_f32(
    const float* __restrict__ A, const float* __restrict__ W,
    const float* __restrict__ bias, float* __restrict__ C,
    int M, int N, int K, int lda, int ldw, int woff)
{
    const int lane = threadIdx.x & 31;
    const int wid  = threadIdx.x >> 5;
    const int tilesN = N >> 4;
    const int tile = blockIdx.x * 8 + wid;   // grids sized so this is always in range
    const int tm = tile / tilesN;
    const int tn = tile - tm * tilesN;
    const int r  = lane & 15;
    const int hh = lane >> 4;

    const float* Arow = A + (size_t)(tm * 16 + r) * lda + 2 * hh;
    const float* Wrow = W + (size_t)(tn * 16 + r) * ldw + woff + 2 * hh;

    v8f acc = {};
    for (int k = 0; k < K; k += 32) {
#pragma unroll
        for (int kk = 0; kk < 32; kk += 4) {
            v2f a = *(const v2f*)(Arow + k + kk);
            v2f b = *(const v2f*)(Wrow + k + kk);
            acc = __builtin_amdgcn_wmma_f32_16x16x4_f32(
                /*neg_a=*/false, a, /*neg_b=*/false, b,
                /*c_mod=*/(short)0, acc, /*reuse_a=*/false, /*reuse_b=*/false);
        }
    }

    const int n = tn * 16 + r;
    const float bv = bias ? bias[n] : 0.0f;
#pragma unroll
    for (int i = 0; i < 8; ++i) {
        const int m = tm * 16 + i + 8 * hh;
        C[(size_t)m * N + n] = acc[i] + bv;
    }
}

// out[((b*T+t)*U+u)*V + v] = le[(b*T+t)*V+v] + lp[(b*U+u)*V+v]
// B=8, T=256, U=64, V=1024 (b_out already folded into lp).
// One float4 per thread; NT stores so the 512MB stream doesn't evict the
// L2-resident le (8MB, re-read 64x) / lp (2MB).
__global__ __launch_bounds__(256) void joiner_bcast_add(
    const float* __restrict__ le, const float* __restrict__ lp,
    float* __restrict__ out)
{
    const size_t i = (size_t)blockIdx.x * 256 + threadIdx.x; // float4 index
    const int    v4   = (int)(i & 255);   // V/4 = 256
    const size_t rest = i >> 8;
    const int    u    = (int)(rest & 63); // U = 64
    const size_t bt   = rest >> 6;        // b*T + t
    const size_t b    = bt >> 8;          // T = 256

    const v4f ev = ((const v4f*)le)[bt * 256 + v4];
    const v4f pv = ((const v4f*)lp)[(b * 64 + u) * 256 + v4];
    v4f o = ev + pv;
    __builtin_nontemporal_store(o, ((v4f*)out) + i);
}

extern "C" void kernel_launch(void* const* d_in, const int* in_sizes, int n_in,
                              void* d_out, int out_size, void* d_ws, size_t ws_size,
                              hipStream_t stream) {
    const float* enc    = (const float*)d_in[0]; // (8,256,512)
    const float* pred   = (const float*)d_in[1]; // (8,64,640)
    const float* W_enc  = (const float*)d_in[2]; // (512,512)
    const float* b_enc  = (const float*)d_in[3]; // (512)
    const float* W_pred = (const float*)d_in[4]; // (512,640)
    const float* b_pred = (const float*)d_in[5]; // (512)
    const float* W_out  = (const float*)d_in[6]; // (1024,1024)
    const float* b_out  = (const float*)d_in[7]; // (1024)
    float* out = (float*)d_out;                  // (8,256,64,1024)

    char* ws = (char*)d_ws;
    float* le = (float*)(ws);                          // 2048*1024*4 = 8 MB
    float* lp = (float*)(ws + ((size_t)8  << 20));     //  512*1024*4 = 2 MB
    float* e  = (float*)(ws + ((size_t)10 << 20));     // 2048*512*4  = 4 MB
    float* p  = (float*)(ws + ((size_t)14 << 20));     //  512*512*4  = 1 MB

    dim3 blk(256);

    // 1) e = enc @ W_enc^T + b_enc        M=2048 N=512  K=512 -> 4096 tiles
    gemm_wmma_f32<<<512, blk, 0, stream>>>(enc, W_enc, b_enc, e,
                                           2048, 512, 512, 512, 512, 0);
    // 2) le = e @ W1^T (W1 = W_out[:, :512])   M=2048 N=1024 K=512 -> 8192 tiles
    gemm_wmma_f32<<<1024, blk, 0, stream>>>(e, W_out, nullptr, le,
                                            2048, 1024, 512, 512, 1024, 0);
    // 3) p = pred @ W_pred^T + b_pred     M=512  N=512  K=640 -> 1024 tiles
    gemm_wmma_f32<<<128, blk, 0, stream>>>(pred, W_pred, b_pred, p,
                                           512, 512, 640, 640, 640, 0);
    // 4) lp = p @ W2^T + b_out (W2 = W_out[:, 512:])  M=512 N=1024 K=512 -> 2048 tiles
    gemm_wmma_f32<<<256, blk, 0, stream>>>(p, W_out, b_out, lp,
                                           512, 1024, 512, 512, 1024, 512);
    // 5) broadcast add: 8*256*64*1024 / 4 = 33,554,432 float4 -> 131072 blocks
    joiner_bcast_add<<<131072, blk, 0, stream>>>(le, lp, out);
}